// MultiScaleDeformableDecoderLayer_44633300140056
// MI455X (gfx1250) — compile-verified
//
#include <hip/hip_runtime.h>
#include <math.h>

// ---------------------------------------------------------------------------
// CDNA5 / gfx1250, wave32. All GEMMs use V_WMMA_F32_16X16X4_F32 (full fp32
// precision, matching the fp32 reference). Fragment layouts per CDNA5 ISA
// 7.12.2:
//   A (16x4 f32):  lane L holds row M=L%16;  vgpr j holds K = j + 2*(L>>4)
//   B (4x16 f32):  lane L holds col N=L%16;  vgpr j holds K = j + 2*(L>>4)
//   C/D (16x16):   lane L holds col N=L%16;  vgpr r holds row M = r + 8*(L>>4)
// ---------------------------------------------------------------------------

typedef float v2f __attribute__((ext_vector_type(2)));
typedef float v8f __attribute__((ext_vector_type(8)));

#define LN_EPS 1e-5f

__device__ __forceinline__ v8f wmma_k4(v2f a, v2f b, v8f c) {
  return __builtin_amdgcn_wmma_f32_16x16x4_f32(
      /*neg_a=*/false, a, /*neg_b=*/false, b,
      /*c_mod=*/(short)0, c, /*reuse_a=*/false, /*reuse_b=*/false);
}

// ---------------------------------------------------------------------------
// 1) Sampling points. pts layout: (scale, b, p, 2) in [0,1] space.
// ---------------------------------------------------------------------------
__global__ void points_kernel(const float* __restrict__ grid0,
                              const float* __restrict__ ref,
                              const float* __restrict__ sq,
                              const float* __restrict__ w1, const float* __restrict__ b1,
                              const float* __restrict__ w2, const float* __restrict__ b2,
                              const float* __restrict__ w3, const float* __restrict__ b3,
                              float* __restrict__ pts) {
  int t = threadIdx.x;  // 256 threads, 1 block
  for (int i = t; i < 16 * 64 * 2; i += 256) {
    pts[i] = grid0[i % 128];  // broadcast uniform grid over batch
  }
  const float* ws[3] = {w1, w2, w3};
  const float* bs[3] = {b1, b2, b3};
  for (int s = 0; s < 3; ++s) {
    for (int i = t; i < 16 * 128; i += 256) {
      int b = i >> 7;
      int j = i & 127;
      const float* wr = ws[s] + (size_t)j * 256;
      const float* xr = sq + (size_t)b * 256;
      float acc = bs[s][j];
      for (int k = 0; k < 256; ++k) acc += xr[k] * wr[k];
      int d = j & 1;
      pts[(((s + 1) * 16 + b) * 64 + (j >> 1)) * 2 + d] = ref[b * 2 + d] + acc;
    }
  }
}

// ---------------------------------------------------------------------------
// 2) Bilinear grid sample, border padding, align_corners=False.
//    pixel coord = clip(p * W - 0.5, 0, W-1). Block = (b,p); 256 threads = C.
// ---------------------------------------------------------------------------
__global__ void grid_sample_kernel(const float* __restrict__ feat,
                                   const float* __restrict__ pos,
                                   const float* __restrict__ pts,  // (B,64,2)
                                   float* __restrict__ S, float* __restrict__ P,
                                   int H, int W, int pbase) {
  int bp = blockIdx.x;
  int b = bp >> 6;
  int p = bp & 63;
  int c = threadIdx.x;
  float px = pts[(size_t)(b * 64 + p) * 2 + 0];
  float py = pts[(size_t)(b * 64 + p) * 2 + 1];
  float x = fminf(fmaxf(px * (float)W - 0.5f, 0.0f), (float)(W - 1));
  float y = fminf(fmaxf(py * (float)H - 0.5f, 0.0f), (float)(H - 1));
  float x0f = floorf(x), y0f = floorf(y);
  int x0 = (int)x0f, y0 = (int)y0f;
  int x1 = min(x0 + 1, W - 1), y1 = min(y0 + 1, H - 1);
  float wx = x - x0f, wy = y - y0f;
  float w00 = (1.f - wx) * (1.f - wy), w01 = wx * (1.f - wy);
  float w10 = (1.f - wx) * wy,         w11 = wx * wy;
  size_t base = ((size_t)b * 256 + c) * (size_t)(H * W);
  int i00 = y0 * W + x0, i01 = y0 * W + x1, i10 = y1 * W + x0, i11 = y1 * W + x1;
  const float* f = feat + base;
  const float* q = pos + base;
  float sf = f[i00] * w00 + f[i01] * w01 + f[i10] * w10 + f[i11] * w11;
  float sp = q[i00] * w00 + q[i01] * w01 + q[i10] * w10 + q[i11] * w11;
  size_t o = ((size_t)b * 256 + (pbase + p)) * 256 + c;
  S[o] = sf;
  P[o] = sp;
}

// ---------------------------------------------------------------------------
// 3) fp32 WMMA linear: out[t][n] = act((A[t]+A2[t]) . W[n] + bias[n]).
//    One wave per 16x64 output tile (4 accumulators share the A fragment).
//    grid = (T/16, N/64), block = 32. HAS_A2/RELU are compile-time.
// ---------------------------------------------------------------------------
template <bool HAS_A2, bool RELU>
__global__ void linear_wmma(const float* __restrict__ A, const float* __restrict__ A2,
                            const float* __restrict__ W, const float* __restrict__ bias,
                            float* __restrict__ out, int K, int N) {
  const int lane = threadIdx.x;           // full wave, EXEC all ones
  const int mrow = blockIdx.x * 16 + (lane & 15);
  const int khalf = (lane >> 4) * 2;      // 0 or 2
  const float* arow = A + (size_t)mrow * K + khalf;
  const float* a2row = HAS_A2 ? A2 + (size_t)mrow * K + khalf : nullptr;
  const int n0 = blockIdx.y * 64;
  const float* wrow[4];
  v8f acc[4];
  const v8f z = {0.f, 0.f, 0.f, 0.f, 0.f, 0.f, 0.f, 0.f};
#pragma unroll
  for (int t = 0; t < 4; ++t) {
    wrow[t] = W + (size_t)(n0 + t * 16 + (lane & 15)) * K + khalf;
    acc[t] = z;
  }
#pragma unroll 2
  for (int k0 = 0; k0 < K; k0 += 4) {
    v2f a;
    a.x = arow[k0];
    a.y = arow[k0 + 1];
    if (HAS_A2) {
      a.x += a2row[k0];
      a.y += a2row[k0 + 1];
    }
#pragma unroll
    for (int t = 0; t < 4; ++t) {
      v2f b;
      b.x = wrow[t][k0];
      b.y = wrow[t][k0 + 1];
      acc[t] = wmma_k4(a, b, acc[t]);
    }
  }
  const int mbase = blockIdx.x * 16 + 8 * (lane >> 4);
#pragma unroll
  for (int t = 0; t < 4; ++t) {
    int ncol = n0 + t * 16 + (lane & 15);
    float bv = bias ? bias[ncol] : 0.0f;
#pragma unroll
    for (int r = 0; r < 8; ++r) {
      float v = acc[t][r] + bv;
      if (RELU) v = fmaxf(v, 0.0f);
      out[(size_t)(mbase + r) * N + ncol] = v;
    }
  }
}

// ---------------------------------------------------------------------------
// 4) Attention scores: Sc[bh][i][j] = scale * <Q[b,i,h*32:], K[b,j,h*32:]>
//    One wave per 16x32 tile (2 accumulators share the Q fragment).
//    grid = (B*H, Lq/16, Lk/32), block = 32.
// ---------------------------------------------------------------------------
__global__ void attn_scores(const float* __restrict__ Q, const float* __restrict__ Km,
                            float* __restrict__ Sc, int Lq, int Lk, float scale) {
  int bh = blockIdx.x;
  int b = bh >> 3;
  int h = bh & 7;
  int lane = threadIdx.x;
  int i = blockIdx.y * 16 + (lane & 15);
  int khalf = (lane >> 4) * 2;
  const float* qrow = Q + ((size_t)b * Lq + i) * 256 + h * 32 + khalf;
  int j0 = blockIdx.z * 32;
  const float* krow0 = Km + ((size_t)b * Lk + j0 + (lane & 15)) * 256 + h * 32 + khalf;
  const float* krow1 = krow0 + (size_t)16 * 256;
  const v8f z = {0.f, 0.f, 0.f, 0.f, 0.f, 0.f, 0.f, 0.f};
  v8f acc0 = z, acc1 = z;
#pragma unroll
  for (int k0 = 0; k0 < 32; k0 += 4) {
    v2f a, b0, b1;
    a.x = qrow[k0];   a.y = qrow[k0 + 1];
    b0.x = krow0[k0]; b0.y = krow0[k0 + 1];
    b1.x = krow1[k0]; b1.y = krow1[k0 + 1];
    acc0 = wmma_k4(a, b0, acc0);
    acc1 = wmma_k4(a, b1, acc1);
  }
  int jc = j0 + (lane & 15);
  int ibase = blockIdx.y * 16 + 8 * (lane >> 4);
  float* srow = Sc + (size_t)bh * Lq * Lk;
#pragma unroll
  for (int r = 0; r < 8; ++r) {
    srow[(size_t)(ibase + r) * Lk + jc] = acc0[r] * scale;
    srow[(size_t)(ibase + r) * Lk + jc + 16] = acc1[r] * scale;
  }
}

// ---------------------------------------------------------------------------
// 5) Row softmax (row length 256 == blockDim).
// ---------------------------------------------------------------------------
__global__ void softmax_rows(float* __restrict__ S) {
  __shared__ float red[256];
  int t = threadIdx.x;
  float* p = S + (size_t)blockIdx.x * 256;
  float v = p[t];
  red[t] = v;
  __syncthreads();
  for (int s = 128; s > 0; s >>= 1) {
    if (t < s) red[t] = fmaxf(red[t], red[t + s]);
    __syncthreads();
  }
  float mx = red[0];
  __syncthreads();
  float e = __expf(v - mx);
  red[t] = e;
  __syncthreads();
  for (int s = 128; s > 0; s >>= 1) {
    if (t < s) red[t] += red[t + s];
    __syncthreads();
  }
  p[t] = e / red[0];
}

// ---------------------------------------------------------------------------
// 6) Attention value matmul: O[b,i,h*32+n] = sum_j P[bh,i,j] * V[b,j,h*32+n]
//    One wave per (i-tile, full d=32): 2 accumulators share the P fragment.
//    grid = (B*H, Lq/16), block = 32.
// ---------------------------------------------------------------------------
__global__ void attn_av(const float* __restrict__ Pm, const float* __restrict__ V,
                        float* __restrict__ O, int Lq, int Lk) {
  int bh = blockIdx.x;
  int b = bh >> 3;
  int h = bh & 7;
  int lane = threadIdx.x;
  int i = blockIdx.y * 16 + (lane & 15);
  int khalf = (lane >> 4) * 2;
  const float* prow = Pm + ((size_t)bh * Lq + i) * Lk + khalf;
  const float* vcol0 = V + (size_t)b * Lk * 256 + h * 32 + (lane & 15)
                         + (size_t)khalf * 256;
  const float* vcol1 = vcol0 + 16;
  const v8f z = {0.f, 0.f, 0.f, 0.f, 0.f, 0.f, 0.f, 0.f};
  v8f acc0 = z, acc1 = z;
#pragma unroll 2
  for (int k0 = 0; k0 < Lk; k0 += 4) {
    v2f a, b0, b1;
    a.x = prow[k0];
    a.y = prow[k0 + 1];
    b0.x = vcol0[(size_t)k0 * 256];
    b0.y = vcol0[(size_t)(k0 + 1) * 256];
    b1.x = vcol1[(size_t)k0 * 256];
    b1.y = vcol1[(size_t)(k0 + 1) * 256];
    acc0 = wmma_k4(a, b0, acc0);
    acc1 = wmma_k4(a, b1, acc1);
  }
  int ncol = h * 32 + (lane & 15);
  int ibase = blockIdx.y * 16 + 8 * (lane >> 4);
#pragma unroll
  for (int r = 0; r < 8; ++r) {
    O[((size_t)b * Lq + ibase + r) * 256 + ncol] = acc0[r];
    O[((size_t)b * Lq + ibase + r) * 256 + ncol + 16] = acc1[r];
  }
}

// ---------------------------------------------------------------------------
// 7) y = LayerNorm(X + R) * g + b over C=256. One block per token.
//    permute=1: token = b*32+l, write to out[(l*16+b)*256+c] (L,B,C order).
// ---------------------------------------------------------------------------
__global__ void residual_ln(const float* __restrict__ X, const float* __restrict__ R,
                            const float* __restrict__ g, const float* __restrict__ be,
                            float* __restrict__ out, int permute) {
  __shared__ float red[256];
  int tok = blockIdx.x;
  int c = threadIdx.x;
  float v = X[(size_t)tok * 256 + c] + R[(size_t)tok * 256 + c];
  red[c] = v;
  __syncthreads();
  for (int s = 128; s > 0; s >>= 1) {
    if (c < s) red[c] += red[c + s];
    __syncthreads();
  }
  float mean = red[0] * (1.0f / 256.0f);
  __syncthreads();
  float d = v - mean;
  red[c] = d * d;
  __syncthreads();
  for (int s = 128; s > 0; s >>= 1) {
    if (c < s) red[c] += red[c + s];
    __syncthreads();
  }
  float var = red[0] * (1.0f / 256.0f);
  float y = d * rsqrtf(var + LN_EPS) * g[c] + be[c];
  size_t o;
  if (permute) {
    int b = tok >> 5, l = tok & 31;        // token order (B,L)
    o = ((size_t)(l * 16 + b)) * 256 + c;  // output order (L,B)
  } else {
    o = (size_t)tok * 256 + c;
  }
  out[o] = y;
}

// ---------------------------------------------------------------------------
// 8) q-prep: (L,B,C) -> (B,L,C); qin = tq + tpe, tqp = tq.
// ---------------------------------------------------------------------------
__global__ void prep_q(const float* __restrict__ tq, const float* __restrict__ tpe,
                       float* __restrict__ qin, float* __restrict__ tqp) {
  int idx = blockIdx.x * 256 + threadIdx.x;  // over (l*B+b)*C+c
  int c = idx & 255;
  int lb = idx >> 8;
  int l = lb / 16, b = lb % 16;
  float a = tq[idx];
  size_t o = ((size_t)(b * 32 + l)) * 256 + c;
  qin[o] = a + tpe[idx];
  tqp[o] = a;
}

// ---------------------------------------------------------------------------
// 9) pooled = mean_l tu(B,L,C); new_ref = clip(ref + sigmoid(pooled[:,:2])-.5);
//    cat = [pooled, sq] (B,512). grid = B, block = 256.
// ---------------------------------------------------------------------------
__global__ void pooled_kernel(const float* __restrict__ tu, const float* __restrict__ sq,
                              const float* __restrict__ ref, float* __restrict__ cat,
                              float* __restrict__ ref_out) {
  int b = blockIdx.x;
  int c = threadIdx.x;
  float s = 0.0f;
  for (int l = 0; l < 32; ++l) s += tu[((size_t)(b * 32 + l)) * 256 + c];
  float pooled = s * (1.0f / 32.0f);
  cat[(size_t)b * 512 + c] = pooled;
  cat[(size_t)b * 512 + 256 + c] = sq[(size_t)b * 256 + c];
  if (c < 2) {
    float sig = 1.0f / (1.0f + __expf(-pooled));
    float nr = ref[b * 2 + c] + sig - 0.5f;
    ref_out[b * 2 + c] = fminf(fmaxf(nr, 0.0f), 1.0f);
  }
}

// ---------------------------------------------------------------------------
// Host-side orchestration
// ---------------------------------------------------------------------------
static void lin(const float* A, const float* A2, const float* W, const float* bias,
                float* O, int T, int K, int N, bool relu, hipStream_t stream) {
  dim3 g(T / 16, N / 64);
  if (A2) {
    linear_wmma<true, false><<<g, 32, 0, stream>>>(A, A2, W, bias, O, K, N);
  } else if (relu) {
    linear_wmma<false, true><<<g, 32, 0, stream>>>(A, nullptr, W, bias, O, K, N);
  } else {
    linear_wmma<false, false><<<g, 32, 0, stream>>>(A, nullptr, W, bias, O, K, N);
  }
}

extern "C" void kernel_launch(void* const* d_in, const int* in_sizes, int n_in,
                              void* d_out, int out_size, void* d_ws, size_t ws_size,
                              hipStream_t stream) {
  (void)in_sizes; (void)n_in; (void)out_size; (void)ws_size;
  const float* tq   = (const float*)d_in[0];   // (L,B,C)
  const float* tpe  = (const float*)d_in[1];
  const float* ref  = (const float*)d_in[2];   // (B,2)
  const float* sq   = (const float*)d_in[3];   // (B,C)
  const float* feats[4] = {(const float*)d_in[4], (const float*)d_in[6],
                           (const float*)d_in[8], (const float*)d_in[10]};
  const float* poss[4]  = {(const float*)d_in[5], (const float*)d_in[7],
                           (const float*)d_in[9], (const float*)d_in[11]};
  const float* grid0 = (const float*)d_in[12];
  const float* ow1 = (const float*)d_in[13]; const float* ob1 = (const float*)d_in[14];
  const float* ow2 = (const float*)d_in[15]; const float* ob2 = (const float*)d_in[16];
  const float* ow3 = (const float*)d_in[17]; const float* ob3 = (const float*)d_in[18];
  const float* sa_wi = (const float*)d_in[19]; const float* sa_bi = (const float*)d_in[20];
  const float* sa_wo = (const float*)d_in[21]; const float* sa_bo = (const float*)d_in[22];
  const float* ca_wi = (const float*)d_in[23]; const float* ca_bi = (const float*)d_in[24];
  const float* ca_wo = (const float*)d_in[25]; const float* ca_bo = (const float*)d_in[26];
  const float* ns_g = (const float*)d_in[27]; const float* ns_b = (const float*)d_in[28];
  const float* n1_g = (const float*)d_in[29]; const float* n1_b = (const float*)d_in[30];
  const float* n2_g = (const float*)d_in[31]; const float* n2_b = (const float*)d_in[32];
  const float* fw1 = (const float*)d_in[33]; const float* fb1 = (const float*)d_in[34];
  const float* fw2 = (const float*)d_in[35]; const float* fb2 = (const float*)d_in[36];
  const float* uw0 = (const float*)d_in[37]; const float* ub0 = (const float*)d_in[38];
  const float* uw1 = (const float*)d_in[39]; const float* ub1 = (const float*)d_in[40];

  float* out = (float*)d_out;  // [text_out 131072 | new_ref 32 | new_sq 4096]
  float* ws = (float*)d_ws;

  const size_t BIG = 4096 * 256;  // 1M floats
  size_t o = 0;
  float* pts = ws + o; o += 8192;
  float* S0  = ws + o; o += BIG;
  float* Pp  = ws + o; o += BIG;
  float* Qb  = ws + o; o += BIG;      // self Q; later cross K
  float* Kb  = ws + o; o += BIG;      // self K; later cross V
  float* Vb  = ws + o; o += BIG;      // self V; later cross Q (uses 131072)
  float* Sc  = ws + o; o += 128ull * 256 * 256;   // scores (self; reused cross)
  float* Ob  = ws + o; o += BIG;      // attn out (self; reused cross)
  float* Xb  = ws + o; o += BIG;      // projection out (SA / CA / F2)
  float* S1  = ws + o; o += BIG;
  float* qin = ws + o; o += 131072;
  float* tqp = ws + o; o += 131072;
  float* tu  = ws + o; o += 131072;
  float* F1  = ws + o; o += 512ull * 2048;
  float* cat = ws + o; o += 8192;
  float* Hb  = ws + o; o += 4096;

  const float scl = 0.17677669529663687f;  // 1/sqrt(32)
  const int HWs[4][2] = {{100, 100}, {50, 50}, {25, 25}, {13, 13}};

  // 1) sampling points
  points_kernel<<<1, 256, 0, stream>>>(grid0, ref, sq, ow1, ob1, ow2, ob2, ow3, ob3, pts);

  // 2) grid-sample all 4 scales into S0/Pp (B,256,C)
  for (int s = 0; s < 4; ++s) {
    grid_sample_kernel<<<16 * 64, 256, 0, stream>>>(
        feats[s], poss[s], pts + (size_t)s * 16 * 64 * 2, S0, Pp,
        HWs[s][0], HWs[s][1], s * 64);
  }

  // 3) self-attention
  lin(S0, Pp,      sa_wi,             sa_bi,       Qb, 4096, 256, 256, false, stream);
  lin(S0, Pp,      sa_wi + 256 * 256, sa_bi + 256, Kb, 4096, 256, 256, false, stream);
  lin(S0, nullptr, sa_wi + 512 * 256, sa_bi + 512, Vb, 4096, 256, 256, false, stream);
  attn_scores<<<dim3(128, 16, 8), 32, 0, stream>>>(Qb, Kb, Sc, 256, 256, scl);
  softmax_rows<<<128 * 256, 256, 0, stream>>>(Sc);
  attn_av<<<dim3(128, 16), 32, 0, stream>>>(Sc, Vb, Ob, 256, 256);
  lin(Ob, nullptr, sa_wo, sa_bo, Xb, 4096, 256, 256, false, stream);
  residual_ln<<<4096, 256, 0, stream>>>(S0, Xb, ns_g, ns_b, S1, 0);

  // 4) cross-attention (Q from text, K/V from updated S)
  prep_q<<<512, 256, 0, stream>>>(tq, tpe, qin, tqp);
  lin(qin, nullptr, ca_wi,             ca_bi,       Vb, 512, 256, 256, false, stream);  // Qc
  lin(S1,  Pp,      ca_wi + 256 * 256, ca_bi + 256, Qb, 4096, 256, 256, false, stream); // Kc
  lin(S1,  nullptr, ca_wi + 512 * 256, ca_bi + 512, Kb, 4096, 256, 256, false, stream); // Vc
  attn_scores<<<dim3(128, 2, 8), 32, 0, stream>>>(Vb, Qb, Sc, 32, 256, scl);
  softmax_rows<<<128 * 32, 256, 0, stream>>>(Sc);
  attn_av<<<dim3(128, 2), 32, 0, stream>>>(Sc, Kb, Ob, 32, 256);
  lin(Ob, nullptr, ca_wo, ca_bo, Xb, 512, 256, 256, false, stream);
  residual_ln<<<512, 256, 0, stream>>>(tqp, Xb, n1_g, n1_b, tu, 0);

  // 5) FFN + final LN -> text_out (written permuted to (L,B,C))
  lin(tu, nullptr, fw1, fb1, F1, 512, 256, 2048, true, stream);
  lin(F1, nullptr, fw2, fb2, Xb, 512, 2048, 256, false, stream);
  residual_ln<<<512, 256, 0, stream>>>(tu, Xb, n2_g, n2_b, out, 1);

  // 6) pooled / ref update / sampling-query update
  pooled_kernel<<<16, 256, 0, stream>>>(tu, sq, ref, cat, out + 131072);
  lin(cat, nullptr, uw0, ub0, Hb, 16, 512, 256, true, stream);
  lin(Hb, nullptr, uw1, ub1, out + 131072 + 32, 16, 256, 256, false, stream);
}